// AggregationLoss_19258633355266
// MI455X (gfx1250) — compile-verified
//
#include <hip/hip_runtime.h>
#include <math.h>

// ---------------------------------------------------------------------------
// AggregationLoss for MI455X (gfx1250, wave32).
//
// Roofline: 157 MB read / 23.3 TB/s  =>  ~6.7 us.  ~46 MFLOP total, so this
// is a pure streaming segmented reduction.  Main kernel: float4/int4 streams,
// per-wave x2 lane-parity-replicated LDS histograms (ds_add_f32, no cross-
// wave contention), block-level fold + global_atomic_add_f32 into 720 floats
// of scratch.  Epilogue: one wave computes 144 per-segment losses and sums
// them through v_wmma_f32_16x16x4_f32 (A=ones; sum over row 0 of D equals
// the sum of every B slot, so the reduction is layout-immune).
// ---------------------------------------------------------------------------

typedef __attribute__((ext_vector_type(2))) float v2f;
typedef __attribute__((ext_vector_type(8))) float v8f;

#define NBATCH   16
#define NCHAN    4
#define NPIX     409600            // 640*640
#define NQUAD_B  102400            // NPIX/4
#define NQUADS   (NBATCH * NQUAD_B)
#define NINST    9
#define NSEG     (NBATCH * NINST)  // 144
#define NFIELD   5                 // 0:A(t-sum) 1:tcard 2:Bk(k-sum) 3:kcard 4:D(overlap)
#define NENT     (NFIELD * NSEG)   // 720
#define WSENT    768               // zeroed scratch entries (covers padded epilogue reads)
#define NCOPIES  16                // 8 waves * 2 lane-parity sub-copies
#define CSTRIDE  737               // odd stride >= 720: scatters LDS banks between copies
#define MAINBLK  1024
#define MAINTHR  256

__global__ void zero_ws_kernel(float* __restrict__ ws) {
  for (int e = threadIdx.x; e < WSENT; e += blockDim.x) ws[e] = 0.0f;
}

__global__ __launch_bounds__(MAINTHR) void agg_hist_kernel(
    const float* __restrict__ pred, const int* __restrict__ tlab,
    const int* __restrict__ klab, float* __restrict__ ws) {
  __shared__ float hist[NCOPIES * CSTRIDE];
  for (int e = threadIdx.x; e < NCOPIES * CSTRIDE; e += MAINTHR) hist[e] = 0.0f;
  __syncthreads();

  const int lane  = threadIdx.x & 31;
  const int wave  = threadIdx.x >> 5;
  const int cbase = (wave * 2 + (lane & 1)) * CSTRIDE;  // private copy per (wave, parity)

  const float4* __restrict__ predq = reinterpret_cast<const float4*>(pred);
  const int4*   __restrict__ tq    = reinterpret_cast<const int4*>(tlab);
  const int4*   __restrict__ kq    = reinterpret_cast<const int4*>(klab);

  auto acc = [&](float s, int t, int k, int segb) {
    const int ts = segb + t;
    const int ks = segb + k;
    __hip_atomic_fetch_add(&hist[cbase + 0 * NSEG + ts], s,
                           __ATOMIC_RELAXED, __HIP_MEMORY_SCOPE_WORKGROUP);
    __hip_atomic_fetch_add(&hist[cbase + 1 * NSEG + ts], 1.0f,
                           __ATOMIC_RELAXED, __HIP_MEMORY_SCOPE_WORKGROUP);
    __hip_atomic_fetch_add(&hist[cbase + 2 * NSEG + ks], s,
                           __ATOMIC_RELAXED, __HIP_MEMORY_SCOPE_WORKGROUP);
    __hip_atomic_fetch_add(&hist[cbase + 3 * NSEG + ks], 1.0f,
                           __ATOMIC_RELAXED, __HIP_MEMORY_SCOPE_WORKGROUP);
    __hip_atomic_fetch_add(&hist[cbase + 4 * NSEG + ks], (t == k) ? s : 0.0f,
                           __ATOMIC_RELAXED, __HIP_MEMORY_SCOPE_WORKGROUP);
  };

  for (int q = blockIdx.x * MAINTHR + threadIdx.x; q < NQUADS;
       q += gridDim.x * MAINTHR) {
    const int b  = q / NQUAD_B;
    const int p4 = q - b * NQUAD_B;
    const float4* pb = predq + (size_t)(b * NCHAN) * NQUAD_B + p4;
    const float4 v0 = pb[0];
    const float4 v1 = pb[1 * NQUAD_B];
    const float4 v2 = pb[2 * NQUAD_B];
    const float4 v3 = pb[3 * NQUAD_B];
    const int4 tl = tq[q];
    const int4 kl = kq[q];
    const int segb = b * NINST;

    const float s0 = v0.x * v0.x + v1.x * v1.x + v2.x * v2.x + v3.x * v3.x;
    const float s1 = v0.y * v0.y + v1.y * v1.y + v2.y * v2.y + v3.y * v3.y;
    const float s2 = v0.z * v0.z + v1.z * v1.z + v2.z * v2.z + v3.z * v3.z;
    const float s3 = v0.w * v0.w + v1.w * v1.w + v2.w * v2.w + v3.w * v3.w;

    acc(s0, tl.x, kl.x, segb);
    acc(s1, tl.y, kl.y, segb);
    acc(s2, tl.z, kl.z, segb);
    acc(s3, tl.w, kl.w, segb);
  }

  __syncthreads();
  // Fold the 16 private copies and push one global atomic per entry per block.
  for (int e = threadIdx.x; e < NENT; e += MAINTHR) {
    float sum = 0.0f;
#pragma unroll
    for (int c2 = 0; c2 < NCOPIES; ++c2) sum += hist[c2 * CSTRIDE + e];
    __hip_atomic_fetch_add(&ws[e], sum, __ATOMIC_RELAXED,
                           __HIP_MEMORY_SCOPE_AGENT);
  }
}

// One wave (32 threads, EXEC all ones).  Computes 144 segment losses, sums
// them on the matrix pipe: D = ones(16x4) x B(4x16) accumulated 3x; row 0 of
// D holds column sums, so sum_{n<16} D[0,n] == sum of all 64 B slots per
// wmma, independent of the exact B register layout.
__global__ __launch_bounds__(32) void agg_final_kernel(
    const float* __restrict__ ws, float* __restrict__ out) {
  const int lane = threadIdx.x;
  float lossv[5];
#pragma unroll
  for (int j = 0; j < 5; ++j) {
    const int i = lane + 32 * j;  // 0..159; all reads stay inside WSENT zeroed scratch
    const float Af  = ws[0 * NSEG + i];
    const float TCf = ws[1 * NSEG + i];
    const float Bkf = ws[2 * NSEG + i];
    const float KCf = ws[3 * NSEG + i];
    const float Df  = ws[4 * NSEG + i];
    const float kc  = (KCf > 0.0f) ? KCf : 1.0f;
    const float tc  = (TCf > 0.0f) ? TCf : 1.0f;
    const float ss  = Af + Bkf / (kc * kc) - 2.0f * Df / kc;
    const float sss = (ss > 0.0f) ? ss : 1.0f;
    const float nrm = sqrtf(sss) - 0.5f;  // SIGMA = 0.5
    const float lv  = log1pf(nrm * nrm) / tc;
    const int inst  = i % NINST;
    const bool valid =
        (i < NSEG) && (inst >= 1) && (TCf > 0.0f) && (KCf > 0.0f) && (ss > 0.0f);
    lossv[j] = valid ? lv : 0.0f;
  }

  v2f onesA;
  onesA.x = 1.0f;
  onesA.y = 1.0f;
  v2f b0, b1, b2;
  b0.x = lossv[0]; b0.y = lossv[1];
  b1.x = lossv[2]; b1.y = lossv[3];
  b2.x = lossv[4]; b2.y = 0.0f;

  v8f c = {};
  c = __builtin_amdgcn_wmma_f32_16x16x4_f32(false, onesA, false, b0, (short)0,
                                            c, false, false);
  c = __builtin_amdgcn_wmma_f32_16x16x4_f32(false, onesA, false, b1, (short)0,
                                            c, false, false);
  c = __builtin_amdgcn_wmma_f32_16x16x4_f32(false, onesA, false, b2, (short)0,
                                            c, false, false);

  float v = (lane < 16) ? c[0] : 0.0f;  // D row 0 lives in VGPR0, lanes 0..15
#pragma unroll
  for (int off = 16; off >= 1; off >>= 1) v += __shfl_xor(v, off, 32);
  if (lane == 0) out[0] = v;
}

extern "C" void kernel_launch(void* const* d_in, const int* in_sizes, int n_in,
                              void* d_out, int out_size, void* d_ws,
                              size_t ws_size, hipStream_t stream) {
  (void)in_sizes; (void)n_in; (void)out_size; (void)ws_size;
  const float* pred = (const float*)d_in[0];
  // d_in[1] (regions_mask), d_in[2] (kernels_mask) are unused by the reference math.
  const int* tlab = (const int*)d_in[3];
  const int* klab = (const int*)d_in[4];
  float* ws  = (float*)d_ws;   // needs WSENT*4 = 3072 bytes
  float* out = (float*)d_out;

  hipLaunchKernelGGL(zero_ws_kernel, dim3(1), dim3(256), 0, stream, ws);
  hipLaunchKernelGGL(agg_hist_kernel, dim3(MAINBLK), dim3(MAINTHR), 0, stream,
                     pred, tlab, klab, ws);
  hipLaunchKernelGGL(agg_final_kernel, dim3(1), dim3(32), 0, stream, ws, out);
}